// ClusteringLayer_53223234732548
// MI455X (gfx1250) — compile-verified
//
#include <hip/hip_runtime.h>
#include <hip/hip_bf16.h>
#include <math.h>

// ---------------- problem constants ----------------
#define NNODES   100000
#define C_IN     128
#define FDIM     116
#define FF       (FDIM * FDIM)        // 13456
#define HDIM     256
#define NCLUS    64
#define EPS_BN   1e-5f
#define EPS_CORR 1e-12f

// segment-sum tiling
#define SS_COLS      128              // columns per block (== blockDim)
#define SS_COLTILES  106              // ceil(13456/128)
#define SS_NTILES    50               // node tiles
#define SS_NODES_PER 2000             // 100000 / 50

typedef float v2f __attribute__((ext_vector_type(2)));
typedef float v8f __attribute__((ext_vector_type(8)));

// ---------------- K0: zero the cov accumulator ----------------
__global__ void zero_kernel(float* __restrict__ p, unsigned int n) {
    unsigned int i = blockIdx.x * blockDim.x + threadIdx.x;
    if (i < n) __builtin_nontemporal_store(0.0f, &p[i]);
}

// ---------------- K1: segment-sum (the bandwidth phase) ----------------
// grid = (SS_COLTILES, SS_NTILES), block = SS_COLS threads.
// LDS accumulator: [C_IN][SS_COLS] f32 = 64 KB -> 5 blocks (20 waves)/WGP.
// Label is uniform per node (scalar load), each thread owns column `tid`
// of every cluster row -> plain LDS read-modify-write, no atomics.
// Streaming loads are non-temporal (5.38 GB read once; keep the 6.9 MB
// accumulator resident in L2 instead). unroll 8 -> 8 outstanding 128B
// loads per wave for memory-level parallelism.
__global__ void __launch_bounds__(SS_COLS)
segsum_kernel(const float* __restrict__ x,
              const int* __restrict__ labels,
              float* __restrict__ cov) {
    __shared__ float acc[C_IN * SS_COLS];   // 65536 bytes

    const int tid = threadIdx.x;
    const int col = blockIdx.x * SS_COLS + tid;
    const bool valid = (col < FF);

    #pragma unroll 8
    for (int c = 0; c < C_IN; ++c) acc[c * SS_COLS + tid] = 0.0f;
    __syncthreads();

    const int n0 = blockIdx.y * SS_NODES_PER;
    int n1 = n0 + SS_NODES_PER;
    if (n1 > NNODES) n1 = NNODES;

    if (valid) {
        const float* xp = x + (size_t)n0 * FF + col;
        #pragma unroll 8
        for (int n = n0; n < n1; ++n) {
            // warm L2 ahead of the stream (global_prefetch_b8, no counters)
            __builtin_prefetch(xp + (size_t)16 * FF, 0, 0);
            int lab = labels[n];                        // wave-uniform -> s_load
            float v = __builtin_nontemporal_load(xp);   // TH=NT streaming read
            acc[lab * SS_COLS + tid] += v;              // ds_load + add + ds_store
            xp += FF;
        }
    }
    __syncthreads();

    if (valid) {
        #pragma unroll 4
        for (int c = 0; c < C_IN; ++c)
            atomicAdd(&cov[(size_t)c * FF + col], acc[c * SS_COLS + tid]);
    }
}

// ---------------- K2: correlation readout ----------------
// one block per cluster; thread f handles row f
__global__ void corr_kernel(const float* __restrict__ cov,
                            float* __restrict__ x_corr) {
    const int c = blockIdx.x;
    const int tid = threadIdx.x;   // 128 threads
    __shared__ float d[FDIM];
    if (tid < FDIM) {
        float dv = cov[(size_t)c * FF + tid * FDIM + tid];
        d[tid] = sqrtf(fmaxf(dv, 0.0f));
    }
    __syncthreads();
    if (tid < FDIM) {
        const float* row = cov + (size_t)c * FF + tid * FDIM;
        const float df = d[tid];
        float s = 0.0f;
        for (int k = 0; k < FDIM; ++k)
            s += row[k] / (df * d[k] + EPS_CORR);
        x_corr[c * FDIM + tid] = s / (float)FDIM;
    }
}

// ---------------- K3/K5: batch-norm stats (training mode, biased var) ----
// grid = Fdim blocks, 128 threads (one per batch row). Emits per-feature
// affine (a, b) with bn(x) = x*a + b.
__global__ void bn_stats_kernel(const float* __restrict__ X,
                                const float* __restrict__ gamma,
                                const float* __restrict__ beta,
                                float* __restrict__ a_out,
                                float* __restrict__ b_out,
                                int Fd) {
    const int f = blockIdx.x;
    const int c = threadIdx.x;     // 0..127 == batch rows
    __shared__ float s1[128];
    __shared__ float s2[128];
    float v = X[c * Fd + f];
    s1[c] = v; s2[c] = v * v;
    __syncthreads();
    for (int off = 64; off > 0; off >>= 1) {
        if (c < off) { s1[c] += s1[c + off]; s2[c] += s2[c + off]; }
        __syncthreads();
    }
    if (c == 0) {
        float mu  = s1[0] * (1.0f / 128.0f);
        float var = s2[0] * (1.0f / 128.0f) - mu * mu;
        float a = gamma[f] * rsqrtf(var + EPS_BN);
        a_out[f] = a;
        b_out[f] = beta[f] - mu * a;
    }
}

// ---------------- K4: GEMM1 via V_WMMA_F32_16X16X4_F32 -------------------
// h[128,256] = relu( bn1(x_corr) @ w1^T + b1 ),  M=128 K=116 N=256.
// One wave per 16x16 output tile; grid = (8 Mtiles, 16 Ntiles).
// A frag (ISA 7.12.2): lane l, reg r -> A[l%16][ (l>>4)*2 + r ]
// B frag (symmetric):  lane l, reg r -> B[ (l>>4)*2 + r ][ l%16 ]
// C/D:                 lane l, reg p -> C[ (l>>4)*8 + p ][ l%16 ]
__global__ void __launch_bounds__(32)
gemm1_wmma(const float* __restrict__ xcorr,
           const float* __restrict__ a1, const float* __restrict__ b1c,
           const float* __restrict__ w1, const float* __restrict__ bias1,
           float* __restrict__ h) {
    const int lane = threadIdx.x;
    const int m0 = blockIdx.x * 16;
    const int n0 = blockIdx.y * 16;
    const int half = lane >> 4;
    const int l16  = lane & 15;

    v8f acc = {};
    #pragma unroll 1
    for (int kt = 0; kt < FDIM / 4; ++kt) {        // 29 steps, K = 116
        const int k0 = kt * 4 + half * 2;
        const int k1 = k0 + 1;
        v2f a, b;
        a.x = xcorr[(m0 + l16) * FDIM + k0] * a1[k0] + b1c[k0];
        a.y = xcorr[(m0 + l16) * FDIM + k1] * a1[k1] + b1c[k1];
        b.x = w1[(n0 + l16) * FDIM + k0];          // B[k][n] = w1[n][k]
        b.y = w1[(n0 + l16) * FDIM + k1];
        acc = __builtin_amdgcn_wmma_f32_16x16x4_f32(
                  false, a, false, b, (short)0, acc, false, false);
    }
    const float bn = bias1[n0 + l16];
    #pragma unroll
    for (int p = 0; p < 8; ++p) {
        const int row = m0 + half * 8 + p;
        h[row * HDIM + n0 + l16] = fmaxf(acc[p] + bn, 0.0f);
    }
}

// ---------------- K6: GEMM2 via V_WMMA_F32_16X16X4_F32 -------------------
// logits[128,64] = bn2(h) @ w2^T + b2,  M=128 K=256 N=64. grid = (8,4).
__global__ void __launch_bounds__(32)
gemm2_wmma(const float* __restrict__ h,
           const float* __restrict__ a2, const float* __restrict__ b2c,
           const float* __restrict__ w2, const float* __restrict__ bias2,
           float* __restrict__ logits) {
    const int lane = threadIdx.x;
    const int m0 = blockIdx.x * 16;
    const int n0 = blockIdx.y * 16;
    const int half = lane >> 4;
    const int l16  = lane & 15;

    v8f acc = {};
    #pragma unroll 1
    for (int kt = 0; kt < HDIM / 4; ++kt) {        // 64 steps, K = 256
        const int k0 = kt * 4 + half * 2;
        const int k1 = k0 + 1;
        v2f a, b;
        a.x = h[(m0 + l16) * HDIM + k0] * a2[k0] + b2c[k0];
        a.y = h[(m0 + l16) * HDIM + k1] * a2[k1] + b2c[k1];
        b.x = w2[(n0 + l16) * HDIM + k0];          // B[k][n] = w2[n][k]
        b.y = w2[(n0 + l16) * HDIM + k1];
        acc = __builtin_amdgcn_wmma_f32_16x16x4_f32(
                  false, a, false, b, (short)0, acc, false, false);
    }
    const float bn = bias2[n0 + l16];
    #pragma unroll
    for (int p = 0; p < 8; ++p) {
        const int row = m0 + half * 8 + p;
        logits[row * NCLUS + n0 + l16] = acc[p] + bn;
    }
}

// ---------------- K7: hard gumbel-softmax (straight-through) -------------
// one block per row (64 threads = one output column each)
__global__ void gumbel_kernel(const float* __restrict__ logits,
                              const float* __restrict__ u,
                              float* __restrict__ out) {
    const int c = blockIdx.x;
    const int j = threadIdx.x;     // 0..63
    __shared__ float red[NCLUS];
    __shared__ int   redi[NCLUS];

    float uu = fminf(fmaxf(u[c * NCLUS + j], 1e-6f), 1.0f - 1e-6f);
    float z  = logits[c * NCLUS + j] - logf(-logf(uu));

    // max
    red[j] = z; __syncthreads();
    for (int off = 32; off > 0; off >>= 1) {
        if (j < off) red[j] = fmaxf(red[j], red[j + off]);
        __syncthreads();
    }
    float zmax = red[0]; __syncthreads();

    // sum of exp
    float e = expf(z - zmax);
    red[j] = e; __syncthreads();
    for (int off = 32; off > 0; off >>= 1) {
        if (j < off) red[j] += red[j + off];
        __syncthreads();
    }
    float s = red[0]; __syncthreads();

    // first argmax of z (== argmax of y_soft)
    red[j] = z; redi[j] = j; __syncthreads();
    for (int off = 32; off > 0; off >>= 1) {
        if (j < off) {
            bool take = (red[j + off] > red[j]) ||
                        (red[j + off] == red[j] && redi[j + off] < redi[j]);
            if (take) { red[j] = red[j + off]; redi[j] = redi[j + off]; }
        }
        __syncthreads();
    }
    const int amax = redi[0];

    float ys = e / s;
    float yh = (j == amax) ? 1.0f : 0.0f;
    out[c * NCLUS + j] = (yh - ys) + ys;   // literal straight-through expr
}

// ---------------- driver ----------------
extern "C" void kernel_launch(void* const* d_in, const int* in_sizes, int n_in,
                              void* d_out, int out_size, void* d_ws, size_t ws_size,
                              hipStream_t stream) {
    const float* x_cov  = (const float*)d_in[0];
    const int*   labels = (const int*)  d_in[1];
    const float* gum    = (const float*)d_in[2];
    const float* w1     = (const float*)d_in[3];
    const float* b1     = (const float*)d_in[4];
    const float* g1     = (const float*)d_in[5];
    const float* be1    = (const float*)d_in[6];
    const float* w2     = (const float*)d_in[7];
    const float* b2     = (const float*)d_in[8];
    const float* g2     = (const float*)d_in[9];
    const float* be2    = (const float*)d_in[10];

    float* out      = (float*)d_out;
    float* cov      = out;                               // [128*13456]
    float* x_corr   = out + (size_t)C_IN * FF;           // [128*116]
    float* mask_out = x_corr + (size_t)C_IN * FDIM;      // [128*64]

    float* ws     = (float*)d_ws;
    float* h      = ws;                 // 32768
    float* logits = ws + 32768;         // 8192
    float* a1     = ws + 40960;         // 116
    float* b1c    = ws + 41088;         // 116
    float* a2     = ws + 41216;         // 256
    float* b2c    = ws + 41472;         // 256

    const unsigned int cov_n = (unsigned int)(C_IN * FF);
    zero_kernel<<<(cov_n + 255) / 256, 256, 0, stream>>>(cov, cov_n);

    segsum_kernel<<<dim3(SS_COLTILES, SS_NTILES), SS_COLS, 0, stream>>>(
        x_cov, labels, cov);

    corr_kernel<<<C_IN, 128, 0, stream>>>(cov, x_corr);

    bn_stats_kernel<<<FDIM, 128, 0, stream>>>(x_corr, g1, be1, a1, b1c, FDIM);
    gemm1_wmma<<<dim3(C_IN / 16, HDIM / 16), 32, 0, stream>>>(
        x_corr, a1, b1c, w1, b1, h);

    bn_stats_kernel<<<HDIM, 128, 0, stream>>>(h, g2, be2, a2, b2c, HDIM);
    gemm2_wmma<<<dim3(C_IN / 16, NCLUS / 16), 32, 0, stream>>>(
        h, a2, b2c, w2, b2, logits);

    gumbel_kernel<<<C_IN, NCLUS, 0, stream>>>(logits, gum, mask_out);
}